// AdaProj_17489106829596
// MI455X (gfx1250) — compile-verified
//
#include <hip/hip_runtime.h>
#include <hip/hip_bf16.h>
#include <math.h>

typedef __attribute__((ext_vector_type(2))) float v2f;
typedef __attribute__((ext_vector_type(8))) float v8f;

#define B_SZ 1024
#define C_SZ 512
#define S_SZ 32
#define E_SZ 128
#define NORM_EPS 1e-12f

// V_WMMA_F32_16X16X4_F32: D = A(16x4,f32) * B(4x16,f32) + C(16x16,f32)
// 8-arg pattern per probe doc: (neg_a, A, neg_b, B, c_mod, C, reuse_a, reuse_b)
static __device__ __forceinline__ v8f wmma_f32(v2f a, v2f b, v8f c) {
  return __builtin_amdgcn_wmma_f32_16x16x4_f32(
      /*neg_a=*/false, a, /*neg_b=*/false, b,
      /*c_mod=*/(short)0, c, /*reuse_a=*/false, /*reuse_b=*/false);
}

// ---------------------------------------------------------------------------
// Row-wise L2 normalize: one 128-thread block per row of length E=128.
// ---------------------------------------------------------------------------
__global__ void __launch_bounds__(128) l2norm_rows_kernel(
    const float* __restrict__ in, float* __restrict__ out) {
  const int row = blockIdx.x;
  const int t = threadIdx.x;  // 0..127
  const float v = in[(size_t)row * E_SZ + t];
  float ss = v * v;
#pragma unroll
  for (int off = 16; off >= 1; off >>= 1) ss += __shfl_xor(ss, off, 32);
  __shared__ float red[4];
  if ((t & 31) == 0) red[t >> 5] = ss;
  __syncthreads();
  const float tot = red[0] + red[1] + red[2] + red[3];
  const float inv = 1.0f / fmaxf(sqrtf(tot), NORM_EPS);
  out[(size_t)row * E_SZ + t] = v * inv;
}

// ---------------------------------------------------------------------------
// Per-class Gram matrix G_c = Wn_c * Wn_c^T  ([32,128] -> [32,32]).
// One block of 4 waves per class; each wave owns one 16x16 tile, K=128 via
// 32x v_wmma_f32_16x16x4_f32 out of LDS.
// ---------------------------------------------------------------------------
__global__ void __launch_bounds__(128) gram_kernel(
    const float* __restrict__ Wn, float* __restrict__ G) {
  const int c = blockIdx.x;
  __shared__ float sW[S_SZ * E_SZ];  // 16 KB
  const float* Wc = Wn + (size_t)c * S_SZ * E_SZ;
  for (int idx = threadIdx.x; idx < S_SZ * E_SZ; idx += 128) sW[idx] = Wc[idx];
  __syncthreads();

  const int wave = threadIdx.x >> 5;
  const int lane = threadIdx.x & 31;
  const int mi = (wave >> 1) << 4;   // 0 or 16
  const int ni = (wave & 1) << 4;    // 0 or 16
  const int r = lane & 15;
  const int grp = lane >> 4;
  const int koff = grp << 1;

  v8f acc = {};
  for (int k0 = 0; k0 < E_SZ; k0 += 4) {
    v2f a, b;
    const float* pa = &sW[(mi + r) * E_SZ + k0 + koff];
    const float* pb = &sW[(ni + r) * E_SZ + k0 + koff];  // B[k][n] = Wn_c[n][k]
    a.x = pa[0]; a.y = pa[1];
    b.x = pb[0]; b.y = pb[1];
    acc = wmma_f32(a, b, acc);
  }
  float* Gc = G + (size_t)c * S_SZ * S_SZ;
#pragma unroll
  for (int i = 0; i < 8; ++i)
    Gc[(mi + i + 8 * grp) * S_SZ + ni + r] = acc[i];
}

// ---------------------------------------------------------------------------
// Fused logits: block = 8 waves = (16 batch rows) x (8 classes, 1/wave).
//   GEMM1: q = xn_tile[16,128] * Wn_c^T[128,32]   (64 WMMAs)
//   GEMM2: t = q[16,32] * G_c[32,32]              (16 WMMAs)
//   logits[b,c] = ||q||^2 / max(sqrt(q.t), 1e-12)
// ---------------------------------------------------------------------------
__global__ void __launch_bounds__(256) logits_kernel(
    const float* __restrict__ xn, const float* __restrict__ Wn,
    const float* __restrict__ G, float* __restrict__ logits) {
  const int btile = blockIdx.x;       // 16 rows per tile
  const int cbase = blockIdx.y << 3;  // 8 classes per block

  __shared__ float sXn[16 * E_SZ];    // 8 KB, shared across the 8 waves
  __shared__ float sCoef[8][16 * 33]; // per-wave padded spill, ~16.5 KB

  for (int idx = threadIdx.x; idx < 16 * E_SZ; idx += 256)
    sXn[idx] = xn[(size_t)btile * 16 * E_SZ + idx];
  __syncthreads();

  const int wave = threadIdx.x >> 5;
  const int lane = threadIdx.x & 31;
  const int c = cbase + wave;
  const int r = lane & 15;
  const int grp = lane >> 4;
  const int koff = grp << 1;

  const float* Wc = Wn + (size_t)c * S_SZ * E_SZ;

  // ---- GEMM1: coef tile, A from LDS, B streamed from L2-resident Wn ----
  v8f q0 = {}, q1 = {};
  for (int k0 = 0; k0 < E_SZ; k0 += 4) {
    v2f a, b0, b1;
    const float* pa = &sXn[r * E_SZ + k0 + koff];
    a.x = pa[0]; a.y = pa[1];
    const float* pb0 = Wc + (size_t)r * E_SZ + k0 + koff;        // n = r
    const float* pb1 = Wc + (size_t)(r + 16) * E_SZ + k0 + koff; // n = r+16
    b0.x = pb0[0]; b0.y = pb0[1];
    b1.x = pb1[0]; b1.y = pb1[1];
    q0 = wmma_f32(a, b0, q0);
    q1 = wmma_f32(a, b1, q1);
  }

  // ---- spill coef (C-layout) to LDS, re-read in A-layout (wave-private) ----
  float* sc = &sCoef[wave][0];
#pragma unroll
  for (int i = 0; i < 8; ++i) {
    sc[(i + 8 * grp) * 33 + r]      = q0[i];
    sc[(i + 8 * grp) * 33 + r + 16] = q1[i];
  }
  asm volatile("s_wait_dscnt 0" ::: "memory");  // wave-internal LDS RAW

  // ---- GEMM2: t = q * G_c ----
  const float* Gc = G + (size_t)c * S_SZ * S_SZ;
  v8f t0 = {}, t1 = {};
  for (int k0 = 0; k0 < S_SZ; k0 += 4) {
    v2f a, b0, b1;
    a.x = sc[r * 33 + k0 + koff];
    a.y = sc[r * 33 + k0 + koff + 1];
    b0.x = Gc[(k0 + koff) * S_SZ + r];
    b0.y = Gc[(k0 + koff + 1) * S_SZ + r];
    b1.x = Gc[(k0 + koff) * S_SZ + r + 16];
    b1.y = Gc[(k0 + koff + 1) * S_SZ + r + 16];
    t0 = wmma_f32(a, b0, t0);
    t1 = wmma_f32(a, b1, t1);
  }

  // ---- per-row reductions over N=32 (16 lanes/half x 2 regs) ----
#pragma unroll
  for (int i = 0; i < 8; ++i) {
    float num = q0[i] * q0[i] + q1[i] * q1[i];  // ||q||^2 partials
    float den = q0[i] * t0[i] + q1[i] * t1[i];  // q^T G q partials
#pragma unroll
    for (int off = 8; off >= 1; off >>= 1) {    // stays within 16-lane half
      num += __shfl_xor(num, off, 32);
      den += __shfl_xor(den, off, 32);
    }
    if (r == 0) {
      const float nrm = sqrtf(fmaxf(den, 0.0f));
      logits[(size_t)(btile * 16 + i + 8 * grp) * C_SZ + c] =
          num / fmaxf(nrm, NORM_EPS);
    }
  }
}

// ---------------------------------------------------------------------------
// Per-row scaled log-softmax NLL: block of 256 threads handles C=512.
// ---------------------------------------------------------------------------
__global__ void __launch_bounds__(256) row_loss_kernel(
    const float* __restrict__ logits, const float* __restrict__ y,
    const float* __restrict__ s_ptr, float* __restrict__ lossb) {
  const int b = blockIdx.x;
  const int t = threadIdx.x;  // 0..255
  const float s = *s_ptr;
  const float l0 = s * logits[(size_t)b * C_SZ + t];
  const float l1 = s * logits[(size_t)b * C_SZ + 256 + t];
  const float tg = y[(size_t)b * C_SZ + t] * l0 +
                   y[(size_t)b * C_SZ + 256 + t] * l1;

  __shared__ float smax[8], ssum[8], stgt[8];
  float mx = fmaxf(l0, l1);
#pragma unroll
  for (int off = 16; off >= 1; off >>= 1) mx = fmaxf(mx, __shfl_xor(mx, off, 32));
  if ((t & 31) == 0) smax[t >> 5] = mx;
  __syncthreads();
  float m = smax[0];
#pragma unroll
  for (int i = 1; i < 8; ++i) m = fmaxf(m, smax[i]);

  float es = expf(l0 - m) + expf(l1 - m);
  float tp = tg;
#pragma unroll
  for (int off = 16; off >= 1; off >>= 1) {
    es += __shfl_xor(es, off, 32);
    tp += __shfl_xor(tp, off, 32);
  }
  if ((t & 31) == 0) { ssum[t >> 5] = es; stgt[t >> 5] = tp; }
  __syncthreads();
  if (t == 0) {
    float sum = 0.0f, tgt = 0.0f;
#pragma unroll
    for (int i = 0; i < 8; ++i) { sum += ssum[i]; tgt += stgt[i]; }
    lossb[b] = m + logf(sum) - tgt;  // logsumexp - target logit
  }
}

// Deterministic single-block mean over B=1024 per-row losses.
__global__ void __launch_bounds__(256) mean_kernel(
    const float* __restrict__ lossb, float* __restrict__ out) {
  const int t = threadIdx.x;
  float v = lossb[t] + lossb[t + 256] + lossb[t + 512] + lossb[t + 768];
#pragma unroll
  for (int off = 16; off >= 1; off >>= 1) v += __shfl_xor(v, off, 32);
  __shared__ float red[8];
  if ((t & 31) == 0) red[t >> 5] = v;
  __syncthreads();
  if (t == 0) {
    float s = 0.0f;
#pragma unroll
    for (int i = 0; i < 8; ++i) s += red[i];
    out[0] = s * (1.0f / (float)B_SZ);
  }
}

// ---------------------------------------------------------------------------
extern "C" void kernel_launch(void* const* d_in, const int* in_sizes, int n_in,
                              void* d_out, int out_size, void* d_ws, size_t ws_size,
                              hipStream_t stream) {
  const float* x = (const float*)d_in[0];  // [B,E]
  const float* y = (const float*)d_in[1];  // [B,C] one-hot
  const float* W = (const float*)d_in[2];  // [C,S,E]
  const float* s = (const float*)d_in[3];  // scalar
  float* out = (float*)d_out;

  float* ws = (float*)d_ws;
  float* xn = ws;                                    // B*E      = 131072
  float* Wn = xn + (size_t)B_SZ * E_SZ;              // C*S*E    = 2097152
  float* G  = Wn + (size_t)C_SZ * S_SZ * E_SZ;       // C*S*S    = 524288
  float* lg = G  + (size_t)C_SZ * S_SZ * S_SZ;       // B*C      = 524288
  float* lb = lg + (size_t)B_SZ * C_SZ;              // B        = 1024

  l2norm_rows_kernel<<<B_SZ, E_SZ, 0, stream>>>(x, xn);
  l2norm_rows_kernel<<<C_SZ * S_SZ, E_SZ, 0, stream>>>(W, Wn);
  gram_kernel<<<C_SZ, 128, 0, stream>>>(Wn, G);
  dim3 grid(B_SZ / 16, C_SZ / 8);
  logits_kernel<<<grid, 256, 0, stream>>>(xn, Wn, G, lg);
  row_loss_kernel<<<B_SZ, 256, 0, stream>>>(lg, y, s, lb);
  mean_kernel<<<1, 256, 0, stream>>>(lb, out);
}